// LightGCN_78013785965002
// MI455X (gfx1250) — compile-verified
//
#include <hip/hip_runtime.h>

// LightGCN: acc = (ego0 + A*ego0 + A^2*ego0 + A^3*ego0) / 4
// A is a 150k x 150k sparse matrix given as 4.8M COO edges (unsorted),
// node features are 64 x fp32. Scatter-SpMM -> hardware f32 atomics into an
// L2-resident (192MB) destination buffer.

namespace {
constexpr int kNumUsers = 100000;
constexpr int kNumItems = 50000;
constexpr int kNodes    = kNumUsers + kNumItems;   // 150000
constexpr int kDim      = 64;
constexpr int kVec4PerRow = kDim / 4;              // 16
constexpr int kNodeVec4 = kNodes * kVec4PerRow;    // 2.4M float4
constexpr int kUserVec4 = kNumUsers * kVec4PerRow; // 1.6M float4
}

// --- scatter SpMM: y[src[e]] += val[e] * x[dst[e]] --------------------------
// 16 lanes per edge; each lane handles one float4 slot of the 64-wide row.
// Placed first in the file so the disassembly snippet shows this hot kernel.
__global__ void lightgcn_spmm(const int*   __restrict__ esrc,
                              const int*   __restrict__ edst,
                              const float* __restrict__ eval,
                              const float* __restrict__ x,
                              float*       __restrict__ y,
                              int nnz) {
  const int tid = blockIdx.x * blockDim.x + threadIdx.x;
  const int sub = tid & 15;   // float4 slot within the row
  const int e   = tid >> 4;   // edge id
  if (e >= nnz) return;

  // Stream-prefetch the edge arrays 4096 edges (16KB) ahead; one prefetch set per
  // block (16 edges = 64B per array, one line's worth) to avoid duplicates.
  if (threadIdx.x == 0) {
    __builtin_prefetch(esrc + e + 4096, 0, 0);
    __builtin_prefetch(edst + e + 4096, 0, 0);
    __builtin_prefetch(eval + e + 4096, 0, 0);
  }

  const int   s = esrc[e];
  const int   d = edst[e];
  const float w = eval[e];

  // Gather one float4 of the source row (global_load_b128; L2-resident).
  const float4 xi = *reinterpret_cast<const float4*>(
      x + (size_t)d * kDim + (size_t)sub * 4);

  float a0 = w * xi.x;
  float a1 = w * xi.y;
  float a2 = w * xi.z;
  float a3 = w * xi.w;

  float* yp = y + (size_t)s * kDim + (size_t)sub * 4;

  // Native non-returning fp32 atomics at device scope (STOREcnt-tracked,
  // fire-and-forget into L2 atomic units). Inline asm guarantees the
  // hardware global_atomic_add_f32 path (no CAS loop).
  asm volatile("global_atomic_add_f32 %0, %1, off scope:SCOPE_DEV"
               :: "v"(yp), "v"(a0) : "memory");
  asm volatile("global_atomic_add_f32 %0, %1, off offset:4 scope:SCOPE_DEV"
               :: "v"(yp), "v"(a1) : "memory");
  asm volatile("global_atomic_add_f32 %0, %1, off offset:8 scope:SCOPE_DEV"
               :: "v"(yp), "v"(a2) : "memory");
  asm volatile("global_atomic_add_f32 %0, %1, off offset:12 scope:SCOPE_DEV"
               :: "v"(yp), "v"(a3) : "memory");
}

// --- init: x_cur = concat(user,item); acc = x_cur; x_next = 0 ---------------
__global__ void lightgcn_init(const float4* __restrict__ user,
                              const float4* __restrict__ item,
                              float4* __restrict__ xcur,
                              float4* __restrict__ xnext,
                              float4* __restrict__ acc) {
  int idx = blockIdx.x * blockDim.x + threadIdx.x;
  if (idx >= kNodeVec4) return;
  float4 v = (idx < kUserVec4) ? user[idx] : item[idx - kUserVec4];
  xcur[idx]  = v;
  acc[idx]   = v;
  xnext[idx] = make_float4(0.f, 0.f, 0.f, 0.f);
}

// --- accumulate layer into acc, optionally zero next scatter target ---------
__global__ void lightgcn_accum(const float4* __restrict__ xin,
                               float4* __restrict__ acc,
                               float4* __restrict__ zbuf,
                               float scale, int do_zero) {
  int idx = blockIdx.x * blockDim.x + threadIdx.x;
  if (idx >= kNodeVec4) return;
  float4 v = xin[idx];
  float4 a = acc[idx];
  a.x = (a.x + v.x) * scale;
  a.y = (a.y + v.y) * scale;
  a.z = (a.z + v.z) * scale;
  a.w = (a.w + v.w) * scale;
  acc[idx] = a;
  if (do_zero) zbuf[idx] = make_float4(0.f, 0.f, 0.f, 0.f);
}

extern "C" void kernel_launch(void* const* d_in, const int* in_sizes, int n_in,
                              void* d_out, int out_size, void* d_ws, size_t ws_size,
                              hipStream_t stream) {
  const float* user = (const float*)d_in[0];
  const float* item = (const float*)d_in[1];
  const int*   esrc = (const int*)d_in[2];
  const int*   edst = (const int*)d_in[3];
  const float* eval = (const float*)d_in[4];
  const int    nnz  = in_sizes[2];

  float* bufA = (float*)d_ws;                       // 38.4 MB
  float* bufB = bufA + (size_t)kNodes * kDim;       // 38.4 MB
  float* acc  = (float*)d_out;

  const dim3 blk(256);
  const dim3 nodeGrid((kNodeVec4 + 255) / 256);
  const long long spmmThreads = (long long)nnz * 16;
  const dim3 spmmGrid((unsigned)((spmmThreads + 255) / 256));

  // ego0: acc = x_cur = concat(user,item); bufB zeroed as first scatter dest.
  lightgcn_init<<<nodeGrid, blk, 0, stream>>>(
      (const float4*)user, (const float4*)item,
      (float4*)bufA, (float4*)bufB, (float4*)acc);

  // layer 1: A -> B ; acc += ego1 ; zero A for layer 2
  lightgcn_spmm<<<spmmGrid, blk, 0, stream>>>(esrc, edst, eval, bufA, bufB, nnz);
  lightgcn_accum<<<nodeGrid, blk, 0, stream>>>(
      (const float4*)bufB, (float4*)acc, (float4*)bufA, 1.0f, 1);

  // layer 2: B -> A ; acc += ego2 ; zero B for layer 3
  lightgcn_spmm<<<spmmGrid, blk, 0, stream>>>(esrc, edst, eval, bufB, bufA, nnz);
  lightgcn_accum<<<nodeGrid, blk, 0, stream>>>(
      (const float4*)bufA, (float4*)acc, (float4*)bufB, 1.0f, 1);

  // layer 3: A -> B ; acc = (acc + ego3) / 4
  lightgcn_spmm<<<spmmGrid, blk, 0, stream>>>(esrc, edst, eval, bufA, bufB, nnz);
  lightgcn_accum<<<nodeGrid, blk, 0, stream>>>(
      (const float4*)bufB, (float4*)acc, (float4*)nullptr, 0.25f, 0);
}